// LocationHistoryEncoder_5875515261483
// MI455X (gfx1250) — compile-verified
//
#include <hip/hip_runtime.h>
#include <hip/hip_bf16.h>
#include <cstdint>
#include <cstddef>

#define TBL 1024  // hash slots per row (>= 2x max distinct keys (512))

// Use the CDNA5 async LDS->global store path for the big zero-fill if the
// toolchain exposes the builtin; otherwise fall back to plain b128 stores.
#if defined(__HIP_DEVICE_COMPILE__) && __has_builtin(__builtin_amdgcn_global_store_async_from_lds_b128)
#define USE_ASYNC_FILL 1
#else
#define USE_ASYNC_FILL 0
#endif

// 16-byte GCC-style int vector: matches the builtin's parameter type exactly
// (diagnostic showed '__attribute__((__vector_size__(4 * sizeof(int)))) int __device__ *').
typedef int v4i __attribute__((vector_size(16)));
typedef __attribute__((address_space(1))) v4i g_v4i;   // global (printed as __device__)
typedef __attribute__((address_space(3))) v4i l_v4i;   // LDS

// ---------------------------------------------------------------------------
// Kernel 1: zero the full (B, N) output. 51.2 MB of stores -> pure HBM
// bandwidth. Async path: every lane replays a 16B zero block from LDS via
// global_store_async_from_lds_b128 (tracked on ASYNCcnt).
// ---------------------------------------------------------------------------
__global__ void zero_fill_kernel(float* __restrict__ out, int n) {
    const int n4 = n >> 2;
    long long tid    = (long long)blockIdx.x * blockDim.x + threadIdx.x;
    long long stride = (long long)gridDim.x * blockDim.x;

#if USE_ASYNC_FILL
    __shared__ v4i zbuf;
    if (threadIdx.x == 0) zbuf = (v4i){0, 0, 0, 0};
    __syncthreads();
    l_v4i* lsrc = (l_v4i*)&zbuf;
    v4i* out4 = (v4i*)out;
    for (long long i = tid; i < n4; i += stride) {
        __builtin_amdgcn_global_store_async_from_lds_b128(
            (g_v4i*)(out4 + i), lsrc, /*offset=*/0, /*cpol=*/0);
    }
    asm volatile("s_wait_asynccnt 0x0" ::: "memory");
#else
    float4* __restrict__ out4 = (float4*)out;
    const float4 z{0.f, 0.f, 0.f, 0.f};
    for (long long i = tid; i < n4; i += stride) out4[i] = z;
#endif

    // tail (n not divisible by 4)
    int rem = n - (n4 << 2);
    if (tid < rem) out[(long long)(n4 << 2) + tid] = 0.0f;
}

// ---------------------------------------------------------------------------
// Kernel 2: one block per batch row. LDS open-addressing hash of the <=512
// distinct locations: CAS claim, integer ds_add for counts, ds_max on
// float-as-uint for the recency scatter-max (all values >= 0 so unsigned
// ordering == float ordering). Then row max_freq via LDS max, and <=512
// scattered stores overwrite the pre-zeroed output. Fully deterministic.
// ---------------------------------------------------------------------------
__global__ __launch_bounds__(1024) void scatter_kernel(
    const int*   __restrict__ loc_seq,
    const int*   __restrict__ mask,
    const float* __restrict__ rwp,
    const float* __restrict__ fwp,
    float*       __restrict__ out,
    int L, int N) {
    __shared__ int      hkey[TBL];
    __shared__ int      hcnt[TBL];
    __shared__ unsigned hval[TBL];
    __shared__ int      smaxcnt;

    const int b = blockIdx.x;
    const int t = threadIdx.x;

    for (int s = t; s < TBL; s += blockDim.x) {
        hkey[s] = -1;
        hcnt[s] = 0;
        hval[s] = 0u;
    }
    if (t == 0) smaxcnt = 0;
    __syncthreads();

    const float rw = rwp[0];
    const float fw = fwp[0];

    const long long base = (long long)b * L + t;
    const int loc = loc_seq[base];
    const int m   = mask[base];
    // recency term: rw^(L-1-t), zeroed for masked timesteps
    const float rec = powf(rw, (float)(L - 1 - t)) * (float)m;

    // ---- insert into hash (linear probing) ----
    unsigned h = ((unsigned)loc * 2654435761u) & (TBL - 1);
    for (;;) {
        int prev = atomicCAS(&hkey[h], -1, loc);
        if (prev == -1 || prev == loc) break;
        h = (h + 1) & (TBL - 1);
    }
    atomicAdd(&hcnt[h], m);                         // scatter-add of validity
    atomicMax(&hval[h], __float_as_uint(rec));      // scatter-max of recency
    __syncthreads();

    // ---- row max frequency ----
    for (int s = t; s < TBL; s += blockDim.x) atomicMax(&smaxcnt, hcnt[s]);
    __syncthreads();

    const float inv = fw / fmaxf((float)smaxcnt, 1.0f);

    // ---- emit touched locations (one writer per distinct key) ----
    for (int s = t; s < TBL; s += blockDim.x) {
        const int k = hkey[s];
        if (k >= 0) {
            out[(long long)b * N + k] =
                __uint_as_float(hval[s]) + (float)hcnt[s] * inv;
        }
    }
}

// ---------------------------------------------------------------------------
extern "C" void kernel_launch(void* const* d_in, const int* in_sizes, int n_in,
                              void* d_out, int out_size, void* d_ws, size_t ws_size,
                              hipStream_t stream) {
    const int*   loc_seq = (const int*)d_in[0];
    const int*   mask    = (const int*)d_in[1];
    const float* rw      = (const float*)d_in[2];
    const float* fw      = (const float*)d_in[3];
    float*       out     = (float*)d_out;

    const int B = 256;                 // reference batch size
    const int L = in_sizes[0] / B;     // 512
    const int N = out_size / B;        // 50000

    // Kernel 1: bandwidth-bound zero fill of the whole output.
    const int threads = 256;           // 8 wave32 waves per block
    int n4 = out_size >> 2;
    int blocks = (n4 + threads - 1) / threads;
    if (blocks > 4096) blocks = 4096;
    if (blocks < 1) blocks = 1;
    zero_fill_kernel<<<blocks, threads, 0, stream>>>(out, out_size);

    // Kernel 2: one 512-thread block per batch row (16 wave32 waves).
    scatter_kernel<<<B, L, 0, stream>>>(loc_seq, mask, rw, fw, out, L, N);
}